// TopKBalancedNoisyGate_13615046328976
// MI455X (gfx1250) — compile-verified
//
#include <hip/hip_runtime.h>
#include <hip/hip_bf16.h>
#include <math.h>

// Problem constants (from reference)
constexpr int TOKENS = 8192;
constexpr int D      = 4096;
constexpr int E      = 16;
constexpr int TOPK   = 4;
constexpr float NOISE_EPS = 0.01f;
constexpr float BALANCE_W = 0.01f;

// d_out flat layout (reference return order)
constexpr int SCORE_OFF = 0;                       // 8192*4 floats
constexpr int BAL_OFF   = TOKENS * TOPK;           // 1 float
constexpr int LOAD_OFF  = BAL_OFF + 1;             // 16 floats
constexpr int IMP_OFF   = LOAD_OFF + E;            // 16 floats
constexpr int IDX_OFF   = IMP_OFF + E;             // 8192*4 floats (indices as float)

// Stage-1 tiling: 4 waves cooperate on one 16-token tile (K-split of D),
// 2 tiles per 256-thread block. Ping-pong groups of U k-steps.
constexpr int KSPLIT = 4;
constexpr int CHUNK  = D / KSPLIT;                 // 1024
constexpr int TILES_PER_BLOCK = 2;
constexpr int U      = 8;                          // k-steps per fragment group
constexpr int STEP   = 4 * U;                      // floats of D per group (32)

typedef float v2f __attribute__((ext_vector_type(2)));
typedef float v8f __attribute__((ext_vector_type(8)));

// ---------------------------------------------------------------------------
// Stage 0: zero the atomically-accumulated stats region of d_out
// ---------------------------------------------------------------------------
__global__ void zero_stats(float* __restrict__ out) {
    int i = threadIdx.x;
    if (i < 1 + 2 * E) out[BAL_OFF + i] = 0.0f;
}

// ---------------------------------------------------------------------------
// Fragment group: U k-steps worth of A (x) and two B (Wg1, Wn) fragments,
// all simultaneously live so the compiler must keep 3*U distinct load
// destinations in flight (no WAR collapse onto 3 registers).
// ---------------------------------------------------------------------------
struct Frag {
    v2f a[U];
    v2f bg[U];
    v2f bn[U];
};

__device__ __forceinline__ void load_group(Frag& f,
                                           const float* __restrict__ xrow,
                                           const float* __restrict__ grow,
                                           const float* __restrict__ nrow,
                                           int k) {
#pragma unroll
    for (int u = 0; u < U; ++u) {
        f.a[u]  = *(const v2f*)(xrow + k + 4 * u);
        f.bg[u] = *(const v2f*)(grow + k + 4 * u);
        f.bn[u] = *(const v2f*)(nrow + k + 4 * u);
    }
}

__device__ __forceinline__ void compute_group(const Frag& f, v8f& accg, v8f& accn) {
#pragma unroll
    for (int u = 0; u < U; ++u) {
        // 8 args: (neg_a, A, neg_b, B, c_mod, C, reuse_a, reuse_b)
        accg = __builtin_amdgcn_wmma_f32_16x16x4_f32(
            false, f.a[u], false, f.bg[u], (short)0, accg, false, false);
        accn = __builtin_amdgcn_wmma_f32_16x16x4_f32(
            false, f.a[u], false, f.bn[u], (short)0, accn, false, false);
    }
}

// ---------------------------------------------------------------------------
// Stage 1: dual projection GEMM via V_WMMA_F32_16X16X4_F32.
//   h_pre = x @ Wg1.T   (pre-tanh)
//   n_pre = x @ Wn.T    (pre-softplus)
// 4-way K-split per tile (2048 waves) + double-buffered fragment groups
// (~24-48 outstanding b64 loads per wave) to cover the 23.3 TB/s
// bandwidth-latency product.
// ---------------------------------------------------------------------------
__global__ void gate_proj_wmma(const float* __restrict__ x,
                               const float* __restrict__ Wg1,
                               const float* __restrict__ Wn,
                               float* __restrict__ h_pre,
                               float* __restrict__ n_pre) {
    const int lane   = threadIdx.x & 31;
    const int wave   = threadIdx.x >> 5;     // 0..7
    const int tIdx   = wave >> 2;            // tile within block: 0..1
    const int ks     = wave & (KSPLIT - 1);  // k-split id: 0..3
    const int tile   = blockIdx.x * TILES_PER_BLOCK + tIdx;
    const int row0   = tile * 16;
    const int half   = lane >> 4;            // 0: lanes 0-15 (K+0..1), 1: lanes 16-31 (K+2..3)
    const int lid    = lane & 15;            // A: token row within tile ; B: expert index
    const int k0     = ks * CHUNK;

    // A fragment source: x[row0+lid][k + 2*half .. +1]
    const float* xrow = x   + (size_t)(row0 + lid) * D + k0 + 2 * half;
    // B fragment source: W[lid][k + 2*half .. +1]  (expert = lid)
    const float* grow = Wg1 + (size_t)lid * D + k0 + 2 * half;
    const float* nrow = Wn  + (size_t)lid * D + k0 + 2 * half;

    v8f accg = {};
    v8f accn = {};

    Frag f0, f1;
    load_group(f0, xrow, grow, nrow, 0);

    // Software-pipelined ping-pong: prefetch the next group while the WMMAs
    // of the current group drain. All addresses stay inside [k0, k0+CHUNK).
#pragma unroll 1
    for (int k = 0; k < CHUNK - 2 * STEP; k += 2 * STEP) {
        load_group(f1, xrow, grow, nrow, k + STEP);
        compute_group(f0, accg, accn);
        load_group(f0, xrow, grow, nrow, k + 2 * STEP);
        compute_group(f1, accg, accn);
    }
    load_group(f1, xrow, grow, nrow, CHUNK - STEP);
    compute_group(f0, accg, accn);   // data at CHUNK - 2*STEP
    compute_group(f1, accg, accn);   // data at CHUNK - STEP

    // Cross-wave K-split reduction through LDS.
    // red[wave][j]    : accg VGPR j     (j = 0..7)
    // red[wave][j+8]  : accn VGPR j
    __shared__ float red[8][16][32];
#pragma unroll
    for (int j = 0; j < 8; ++j) {
        red[wave][j][lane]     = accg[j];
        red[wave][j + 8][lane] = accn[j];
    }
    __syncthreads();

    if (ks == 0) {
        const int w0 = wave;  // == tIdx * KSPLIT
#pragma unroll
        for (int j = 0; j < 8; ++j) {
            float g = red[w0][j][lane]     + red[w0 + 1][j][lane]
                    + red[w0 + 2][j][lane] + red[w0 + 3][j][lane];
            float n = red[w0][j + 8][lane]     + red[w0 + 1][j + 8][lane]
                    + red[w0 + 2][j + 8][lane] + red[w0 + 3][j + 8][lane];
            // C/D layout: VGPR j -> token row (j + 8*half), expert = lid
            const int tok = row0 + j + 8 * half;
            h_pre[(size_t)tok * E + lid] = g;
            n_pre[(size_t)tok * E + lid] = n;
        }
    }
}

// ---------------------------------------------------------------------------
// Stage 2: per-token epilogue. One thread per token; all E=16 state in
// registers (fully unrolled). LDS reduction for importance/load.
// ---------------------------------------------------------------------------
__global__ void gate_finish(const float* __restrict__ h_pre,
                            const float* __restrict__ n_pre,
                            const float* __restrict__ noise,
                            const float* __restrict__ Wg2,
                            float* __restrict__ out) {
    __shared__ float s_imp[E];
    __shared__ float s_load[E];
    if (threadIdx.x < E) { s_imp[threadIdx.x] = 0.0f; s_load[threadIdx.x] = 0.0f; }
    __syncthreads();

    const int t = blockIdx.x * blockDim.x + threadIdx.x;

    // h = tanh(x @ Wg1.T)
    float h[E];
#pragma unroll
    for (int j = 0; j < E; ++j) h[j] = tanhf(h_pre[(size_t)t * E + j]);

    // logits_gate = h @ Wg2.T  (Wg2 is uniform -> scalar-cached loads)
    float lg[E];
#pragma unroll
    for (int e = 0; e < E; ++e) {
        float acc = 0.0f;
#pragma unroll
        for (int j = 0; j < E; ++j) acc = fmaf(Wg2[e * E + j], h[j], acc);
        lg[e] = acc;
    }

    // noise_control = softplus(x @ Wn.T) + eps ; noisy logits
    float nc[E], ln[E], logit[E];
#pragma unroll
    for (int e = 0; e < E; ++e) {
        float v  = n_pre[(size_t)t * E + e];
        float sp = (v > 20.0f) ? v : log1pf(expf(v));
        nc[e]    = sp + NOISE_EPS;
        ln[e]    = noise[(size_t)t * E + e] * nc[e];
        logit[e] = lg[e] + ln[e];
    }

    // top-5 selection (strict > => lowest index wins ties, matching lax.top_k)
    float val[TOPK + 1];
    int   idx[TOPK + 1];
    unsigned used = 0u;
#pragma unroll
    for (int s = 0; s < TOPK + 1; ++s) {
        float best = -INFINITY;
        int   bi   = 0;
#pragma unroll
        for (int e = 0; e < E; ++e) {
            bool ok = (((used >> e) & 1u) == 0u) && (logit[e] > best);
            best = ok ? logit[e] : best;
            bi   = ok ? e : bi;
        }
        used |= (1u << bi);
        val[s] = best;
        idx[s] = bi;
    }

    // softmax over top-4 (val[0] is the max)
    float ex[TOPK];
    float sum = 0.0f;
#pragma unroll
    for (int s = 0; s < TOPK; ++s) { ex[s] = expf(val[s] - val[0]); sum += ex[s]; }
    const float inv = 1.0f / sum;
#pragma unroll
    for (int s = 0; s < TOPK; ++s) {
        const float sc = ex[s] * inv;
        out[SCORE_OFF + t * TOPK + s] = sc;
        out[IDX_OFF   + t * TOPK + s] = (float)idx[s];
        atomicAdd(&s_imp[idx[s]], sc);
    }

    // load probabilities via ndtr(z) = 0.5*erfc(-z/sqrt(2))
    const float thr_in  = val[TOPK];      // (K+1)-th largest
    const float thr_out = val[TOPK - 1];  // K-th largest
#pragma unroll
    for (int e = 0; e < E; ++e) {
        const bool  in = ln[e] > thr_in;
        const float z  = (lg[e] - (in ? thr_in : thr_out)) / nc[e];
        const float p  = 0.5f * erfcf(-z * 0.70710678118654752f);
        atomicAdd(&s_load[e], p);
    }

    __syncthreads();
    if (threadIdx.x < E) {
        atomicAdd(&out[LOAD_OFF + threadIdx.x], s_load[threadIdx.x]);
        atomicAdd(&out[IMP_OFF  + threadIdx.x], s_imp[threadIdx.x]);
    }
}

// ---------------------------------------------------------------------------
// Stage 3: balance loss = 0.01 * (cv2(importance) + cv2(load)), ddof=1
// ---------------------------------------------------------------------------
__device__ float cv_squared16(const float* v) {
    float s = 0.0f;
#pragma unroll
    for (int i = 0; i < E; ++i) s += v[i];
    const float mean = s / (float)E;
    float q = 0.0f;
#pragma unroll
    for (int i = 0; i < E; ++i) { const float d = v[i] - mean; q += d * d; }
    const float var = q / (float)(E - 1);
    return var / (mean * mean + 1e-10f);
}

__global__ void balance_kernel(float* __restrict__ out) {
    if (threadIdx.x != 0 || blockIdx.x != 0) return;
    float imp[E], ld[E];
#pragma unroll
    for (int i = 0; i < E; ++i) { ld[i] = out[LOAD_OFF + i]; imp[i] = out[IMP_OFF + i]; }
    out[BAL_OFF] = (cv_squared16(imp) + cv_squared16(ld)) * BALANCE_W;
}

// ---------------------------------------------------------------------------
extern "C" void kernel_launch(void* const* d_in, const int* in_sizes, int n_in,
                              void* d_out, int out_size, void* d_ws, size_t ws_size,
                              hipStream_t stream) {
    (void)in_sizes; (void)n_in; (void)out_size; (void)ws_size;

    const float* x     = (const float*)d_in[0];
    const float* noise = (const float*)d_in[1];
    const float* Wg1   = (const float*)d_in[2];
    const float* Wg2   = (const float*)d_in[3];
    const float* Wn    = (const float*)d_in[4];
    float*       out   = (float*)d_out;

    float* ws    = (float*)d_ws;
    float* h_pre = ws;                          // 8192*16 fp32
    float* n_pre = ws + (size_t)TOKENS * E;     // 8192*16 fp32

    zero_stats<<<1, 64, 0, stream>>>(out);

    // 512 tiles of 16 tokens; 2 tiles per block, 4 K-split waves per tile
    gate_proj_wmma<<<(TOKENS / 16) / TILES_PER_BLOCK, 256, 0, stream>>>(
        x, Wg1, Wn, h_pre, n_pre);

    gate_finish<<<TOKENS / 256, 256, 0, stream>>>(h_pre, n_pre, noise, Wg2, out);

    balance_kernel<<<1, 1, 0, stream>>>(out);
}